// SimilarityModel_24584392802842
// MI455X (gfx1250) — compile-verified
//
#include <hip/hip_runtime.h>

typedef __attribute__((ext_vector_type(2))) float v2f;
typedef __attribute__((ext_vector_type(8))) float v8f;

namespace {
constexpr int kB      = 64;     // songs
constexpr int kF      = 128;    // mel bins
constexpr int kT      = 128;    // frames
constexpr int kNPos   = 16384;  // positive rows (sorted segments)
constexpr int kNumNeg = 256;    // negatives per song
constexpr int kTextF  = 128;    // text feature dim
constexpr int kD      = 64;     // shared embedding dim

constexpr int kRowsPerBlock = 32;   // text rows handled by one 256-thread block
constexpr int kLdsStride    = 66;   // 64 + 2 pad: stride-2 bank spread for K-column reads
}

// D = A(16x4, fp32) * B(4x16, fp32) + C(16x16, fp32) -> v_wmma_f32_16x16x4_f32
__device__ __forceinline__ v8f wmma4(v2f a, v2f b, v8f c) {
  return __builtin_amdgcn_wmma_f32_16x16x4_f32(
      /*neg_a=*/false, a, /*neg_b=*/false, b,
      /*c_mod=*/(short)0, c, /*reuse_a=*/false, /*reuse_b=*/false);
}

// ---------------------------------------------------------------------------
// Kernel 1: audio encoder. Per song b: A[b] = Wa(64x128) @ spec[b](128x128).
// One wave per 16x16 output tile; 2048 waves total. A (2MB) stays in L2.
// ---------------------------------------------------------------------------
__global__ void __launch_bounds__(256) k_audio_encode(
    const float* __restrict__ spec, const float* __restrict__ Wa,
    float* __restrict__ A) {
  const int lane = threadIdx.x & 31;
  const int gw   = blockIdx.x * 8 + (threadIdx.x >> 5);  // 0..2047
  const int b    = gw >> 5;                              // 32 tiles per song
  const int mt   = (gw >> 3) & 3;                        // 4 tiles over D
  const int nt   = gw & 7;                               // 8 tiles over T
  const int lo = lane & 15, hi = lane >> 4;
  const int m = mt * 16 + lo;          // output row (d)
  const int t = nt * 16 + lo;          // output col (frame)
  const float* sp = spec + b * (kF * kT);

  v8f acc = {};
#pragma unroll 4
  for (int k0 = 0; k0 < kF; k0 += 4) {
    const int ka = k0 + hi * 2;
    v2f a, bb;
    a.x  = Wa[m * kF + ka];
    a.y  = Wa[m * kF + ka + 1];
    bb.x = sp[ka * kT + t];
    bb.y = sp[(ka + 1) * kT + t];
    acc = wmma4(a, bb, acc);
  }
#pragma unroll
  for (int r = 0; r < 8; ++r)
    A[b * (kD * kT) + (mt * 16 + hi * 8 + r) * kT + t] = acc[r];
}

// ---------------------------------------------------------------------------
// Kernel 2 (fused): text projection + similarity scoring for a 32-row block.
//   Phase 1: P = X[n0:n0+32,:] @ Wt^T  (8 waves -> 8 WMMA tiles) -> LDS
//   Phase 2: out[n] = P[n] . A[seg(n)] (16 WMMA tiles, 2 per wave),
//            A-fragments from LDS, B-fragments from L2-resident A.
// seg == nullptr  => negatives: seg(n) = n / 256 (always tile-uniform).
// seg != nullptr  => positives: sorted ids; tile uniform iff first==last,
//                    boundary tiles (<=63 of 1024) take a scalar fallback.
// ---------------------------------------------------------------------------
__global__ void __launch_bounds__(256) k_project_score(
    const float* __restrict__ X,    // [16384,128] pos or neg text features
    const float* __restrict__ Wt,   // [64,128]
    const float* __restrict__ A,    // [64,64,128] audio embeddings
    const int* __restrict__ seg,    // sorted segment ids, or nullptr
    float* __restrict__ out) {      // [16384,128]
  __shared__ float ldsP[kRowsPerBlock * kLdsStride];

  const int lane = threadIdx.x & 31;
  // Scalarize the wave id so per-wave control flow below is SALU (EXEC
  // provably all-1s around WMMA, as the ISA requires).
  const int wave = __builtin_amdgcn_readfirstlane(threadIdx.x >> 5);  // 0..7
  const int lo = lane & 15, hi = lane >> 4;
  const int n0 = blockIdx.x * kRowsPerBlock;

  // ---- Phase 1: projection tiles (2 row-groups x 4 d-tiles) ----
  {
    const int rt = wave >> 2;              // row-group 0/1
    const int dt = wave & 3;               // d-tile 0..3
    const int row  = n0 + rt * 16 + lo;
    const int dcol = dt * 16 + lo;
    v8f acc = {};
#pragma unroll 4
    for (int k0 = 0; k0 < kTextF; k0 += 4) {
      const int ka = k0 + hi * 2;
      v2f a, bb;
      a.x  = X[row * kTextF + ka];
      a.y  = X[row * kTextF + ka + 1];
      bb.x = Wt[dcol * kTextF + ka];       // B[k][n] = Wt[n][k]
      bb.y = Wt[dcol * kTextF + ka + 1];
      acc = wmma4(a, bb, acc);
    }
#pragma unroll
    for (int r = 0; r < 8; ++r)
      ldsP[(rt * 16 + hi * 8 + r) * kLdsStride + dt * 16 + lo] = acc[r];
  }
  __syncthreads();

  // ---- Phase 2: scoring tiles (2 row-groups x 8 t-tiles), 2 per wave ----
  for (int tile = wave; tile < 16; tile += 8) {
    const int rg = tile >> 3;              // local row-group
    const int nt = tile & 7;               // t-tile
    const int rbase = rg * 16;             // local row base in LDS
    const int gn0   = n0 + rbase;          // global row base
    const int t0    = nt * 16;

    int s_first, s_last;
    if (seg) { s_first = seg[gn0]; s_last = seg[gn0 + 15]; }
    else     { s_first = s_last = gn0 / kNumNeg; }

    if (s_first == s_last) {
      const float* Ab = A + s_first * (kD * kT);
      v8f acc = {};
#pragma unroll 4
      for (int k0 = 0; k0 < kD; k0 += 4) {
        const int ka = k0 + hi * 2;
        v2f a, bb;
        a.x  = ldsP[(rbase + lo) * kLdsStride + ka];
        a.y  = ldsP[(rbase + lo) * kLdsStride + ka + 1];
        bb.x = Ab[ka * kT + t0 + lo];
        bb.y = Ab[(ka + 1) * kT + t0 + lo];
        acc = wmma4(a, bb, acc);
      }
#pragma unroll
      for (int r = 0; r < 8; ++r)
        out[(gn0 + hi * 8 + r) * kT + t0 + lo] = acc[r];
    } else {
      // segment-boundary tile: per-lane dot products, P read from LDS
#pragma unroll
      for (int r = 0; r < 8; ++r) {
        const int lrow = rbase + hi * 8 + r;
        const int s = seg[n0 + lrow];
        const float* Ab = A + s * (kD * kT) + t0 + lo;
        float acc = 0.f;
        for (int d = 0; d < kD; ++d)
          acc += ldsP[lrow * kLdsStride + d] * Ab[d * kT];
        out[(n0 + lrow) * kT + t0 + lo] = acc;
      }
    }
  }
}

// ---------------------------------------------------------------------------
extern "C" void kernel_launch(void* const* d_in, const int* in_sizes, int n_in,
                              void* d_out, int out_size, void* d_ws, size_t ws_size,
                              hipStream_t stream) {
  const float* spec = (const float*)d_in[0];   // [64,128,128]
  const float* pos  = (const float*)d_in[1];   // [16384,128]
  const float* neg  = (const float*)d_in[2];   // [16384,128]
  const int*   pseg = (const int*)d_in[3];     // [16384] sorted
  const float* Wa   = (const float*)d_in[4];   // [64,128]
  const float* Wt   = (const float*)d_in[5];   // [64,128]
  float* out = (float*)d_out;                  // PA(16384*128) ++ NA(16384*128)

  // Workspace: only A (64*64*128 fp32 = 2MB), fully L2-resident.
  float* A = (float*)d_ws;

  // audio encoder: 64 songs * (4x8) tiles = 2048 waves
  k_audio_encode<<<(kB * (kD / 16) * (kT / 16)) / 8, 256, 0, stream>>>(spec, Wa, A);
  // fused project+score: 16384/32 = 512 blocks each
  k_project_score<<<kNPos / kRowsPerBlock, 256, 0, stream>>>(
      pos, Wt, A, pseg, out);
  k_project_score<<<kNPos / kRowsPerBlock, 256, 0, stream>>>(
      neg, Wt, A, nullptr, out + kNPos * kT);
}